// SimpleGAT_63118839382176
// MI455X (gfx1250) — compile-verified
//
#include <hip/hip_runtime.h>

#define NNODES 100000
#define NEDGES 1600000
#define ETOT   (NEDGES + NNODES)   // 1,700,000 with self-loops
#define F_IN   128
#define HC1    128                 // HEADS*HID
#define HC2    64                  // HEADS*CLS
#define HEADS  4
#define NEG_SLOPE 0.2f
#define BN_EPS 1e-5f

typedef __attribute__((ext_vector_type(2))) float v2f;
typedef __attribute__((ext_vector_type(8))) float v8f;

// ---------------------------------------------------------------------------
// FP32 GEMM via V_WMMA_F32_16X16X4_F32, compile-time K/NOUT.
// One wave owns a 16 x NOUT output strip: A (16x4 slice of X) is loaded once
// per k-step and reused across all NOUT/16 column tiles (8 independent WMMAs
// per A-load for NOUT=128).  Layouts per ISA:
//   A 16x4 : lanes 0-15 row M, vgpr0/1 = K0/K1; lanes 16-31 = K2/K3
//   B 4x16 : lanes 0-15 col N, vgpr r = row r; lanes 16-31 = row r+2
//   C/D    : vgpr r -> row half*8 + r, col = lane&15
// ---------------------------------------------------------------------------
template <int K, int NOUT>
__global__ __launch_bounds__(128)
void gemm_wmma_f32(const float* __restrict__ X, const float* __restrict__ W,
                   float* __restrict__ Out, int M) {
  constexpr int NT = NOUT / 16;              // column tiles per strip
  int strip = blockIdx.x * 4 + threadIdx.y;  // 16-row strip index
  if (strip >= (M >> 4)) return;             // wave-uniform exit (EXEC all-1)
  int lane = threadIdx.x;                    // 0..31 (wave32)
  int half = lane >> 4;                      // 0: K0/K1 , 1: K2/K3
  int l15  = lane & 15;

  const float* __restrict__ Xp = X + (size_t)((strip << 4) + l15) * K + half * 2;
  const float* __restrict__ Wp = W + half * 2 * NOUT + l15;

  v8f acc[NT];
#pragma unroll
  for (int t = 0; t < NT; ++t) acc[t] = (v8f){};

  for (int k = 0; k < K; k += 4) {
    v2f a;
    a.x = Xp[k];
    a.y = Xp[k + 1];
#pragma unroll
    for (int t = 0; t < NT; ++t) {
      v2f b;
      b.x = Wp[k * NOUT + t * 16];
      b.y = Wp[(k + 1) * NOUT + t * 16];
      acc[t] = __builtin_amdgcn_wmma_f32_16x16x4_f32(false, a, false, b,
                                                     (short)0, acc[t], false, false);
    }
  }

  float* __restrict__ Op = Out + (size_t)((strip << 4) + half * 8) * NOUT + l15;
#pragma unroll
  for (int t = 0; t < NT; ++t)
#pragma unroll
    for (int r = 0; r < 8; ++r) Op[r * NOUT + t * 16] = acc[t][r];
}

// ---------------------------------------------------------------------------
// Order-preserving float <-> uint key for atomicMax-based segment max.
// ---------------------------------------------------------------------------
__device__ __forceinline__ unsigned int f2key(float f) {
  unsigned int b = __float_as_uint(f);
  return (b & 0x80000000u) ? ~b : (b | 0x80000000u);
}
__device__ __forceinline__ float key2f(unsigned int k) {
  unsigned int b = (k & 0x80000000u) ? (k & 0x7fffffffu) : ~k;
  return __uint_as_float(b);
}

// ---------------------------------------------------------------------------
// Pass 1: per-edge per-head attention logits + scatter max over dst.
// One wave per edge; lane handles channels {lane, lane+32, ...} (coalesced).
// ---------------------------------------------------------------------------
template <int C>
__global__ __launch_bounds__(256)
void edge_logits_kernel(const float* __restrict__ xl, const float* __restrict__ xr,
                        const int* __restrict__ ei, const float* __restrict__ att,
                        float* __restrict__ logits, unsigned int* __restrict__ maxkey) {
  const int HC = HEADS * C;
  int e = blockIdx.x * (blockDim.x >> 5) + (threadIdx.x >> 5);
  if (e >= ETOT) return;
  int lane = threadIdx.x & 31;
  int s, d;
  if (e < NEDGES) { s = ei[e]; d = ei[NEDGES + e]; } else { s = e - NEDGES; d = s; }

  float hs[HEADS] = {0.f, 0.f, 0.f, 0.f};
#pragma unroll
  for (int j = 0; j < HC / 32; ++j) {
    int ch = lane + 32 * j;
    float v = xl[(size_t)s * HC + ch] + xr[(size_t)d * HC + ch];
    v = v > 0.f ? v : v * NEG_SLOPE;
    float c = att[ch] * v;
    int h = ch / C;
#pragma unroll
    for (int hh = 0; hh < HEADS; ++hh) hs[hh] += (h == hh) ? c : 0.f;
  }
#pragma unroll
  for (int hh = 0; hh < HEADS; ++hh) {
#pragma unroll
    for (int off = 16; off > 0; off >>= 1) hs[hh] += __shfl_xor(hs[hh], off, 32);
  }
  if (lane == 0) {
#pragma unroll
    for (int h = 0; h < HEADS; ++h) {
      float lg = hs[h];
      logits[(size_t)e * HEADS + h] = lg;
      atomicMax(&maxkey[(size_t)d * HEADS + h], f2key(lg));
    }
  }
}

// ---------------------------------------------------------------------------
// Pass 2: ex = exp(logit - max[dst]); denom[dst] += ex  (in-place on logits).
// ---------------------------------------------------------------------------
__global__ __launch_bounds__(256)
void edge_exp_kernel(float* __restrict__ logits, const int* __restrict__ ei,
                     const unsigned int* __restrict__ maxkey, float* __restrict__ denom) {
  int idx = blockIdx.x * blockDim.x + threadIdx.x;
  if (idx >= ETOT * HEADS) return;
  int e = idx >> 2, h = idx & 3;
  int d = (e < NEDGES) ? ei[NEDGES + e] : e - NEDGES;
  float m  = key2f(maxkey[(size_t)d * HEADS + h]);
  float ex = expf(logits[idx] - m);
  logits[idx] = ex;
  atomicAdd(&denom[(size_t)d * HEADS + h], ex);
}

// ---------------------------------------------------------------------------
// Pass 3: out[dst] += xl[src] * alpha  (wave per edge, coalesced atomics).
// ---------------------------------------------------------------------------
template <int C>
__global__ __launch_bounds__(256)
void edge_agg_kernel(const float* __restrict__ xl, const float* __restrict__ ex,
                     const float* __restrict__ denom, const int* __restrict__ ei,
                     float* __restrict__ out) {
  const int HC = HEADS * C;
  int e = blockIdx.x * (blockDim.x >> 5) + (threadIdx.x >> 5);
  if (e >= ETOT) return;
  int lane = threadIdx.x & 31;
  int s, d;
  if (e < NEDGES) { s = ei[e]; d = ei[NEDGES + e]; } else { s = e - NEDGES; d = s; }
  float a0 = ex[(size_t)e * 4 + 0] / denom[(size_t)d * 4 + 0];
  float a1 = ex[(size_t)e * 4 + 1] / denom[(size_t)d * 4 + 1];
  float a2 = ex[(size_t)e * 4 + 2] / denom[(size_t)d * 4 + 2];
  float a3 = ex[(size_t)e * 4 + 3] / denom[(size_t)d * 4 + 3];
#pragma unroll
  for (int j = 0; j < HC / 32; ++j) {
    int ch = lane + 32 * j;
    int h  = ch / C;
    float a = (h == 0) ? a0 : (h == 1) ? a1 : (h == 2) ? a2 : a3;
    atomicAdd(&out[(size_t)d * HC + ch], xl[(size_t)s * HC + ch] * a);
  }
}

// ---------------------------------------------------------------------------
// BatchNorm helpers (population stats over nodes) + ReLU.
// ---------------------------------------------------------------------------
__global__ __launch_bounds__(256)
void bn_reduce_kernel(const float* __restrict__ h, float* __restrict__ sum,
                      float* __restrict__ sumsq) {
  int idx = blockIdx.x * blockDim.x + threadIdx.x;
  int stride = gridDim.x * blockDim.x;        // multiple of 128
  int ch = threadIdx.x & (HC1 - 1);
  float s = 0.f, ss = 0.f;
  for (int i = idx; i < NNODES * HC1; i += stride) {
    float v = h[i];
    s += v; ss += v * v;
  }
  atomicAdd(&sum[ch], s);
  atomicAdd(&sumsq[ch], ss);
}

__global__ __launch_bounds__(256)
void bn_apply_kernel(float* __restrict__ h, const float* __restrict__ sum,
                     const float* __restrict__ sumsq, const float* __restrict__ gamma,
                     const float* __restrict__ beta) {
  int i = blockIdx.x * blockDim.x + threadIdx.x;
  if (i >= NNODES * HC1) return;
  int ch = i & (HC1 - 1);
  const float invN = 1.0f / (float)NNODES;
  float mu  = sum[ch] * invN;
  float var = sumsq[ch] * invN - mu * mu;
  float v = (h[i] - mu) * rsqrtf(var + BN_EPS) * gamma[ch] + beta[ch];
  h[i] = v > 0.f ? v : 0.f;                    // ReLU (dropout p=0 is identity)
}

__global__ __launch_bounds__(256)
void fill_u32_kernel(unsigned int* __restrict__ p, unsigned int v, int n) {
  int i = blockIdx.x * blockDim.x + threadIdx.x;
  if (i < n) p[i] = v;
}

__global__ __launch_bounds__(256)
void init_bias_kernel(float* __restrict__ out, const float* __restrict__ bias,
                      int total, int mask) {
  int i = blockIdx.x * blockDim.x + threadIdx.x;
  if (i < total) out[i] = bias[i & mask];
}

// ---------------------------------------------------------------------------
extern "C" void kernel_launch(void* const* d_in, const int* in_sizes, int n_in,
                              void* d_out, int out_size, void* d_ws, size_t ws_size,
                              hipStream_t stream) {
  const float* x     = (const float*)d_in[0];
  const int*   ei    = (const int*)  d_in[1];   // row0: src[E], row1: dst[E]
  const float* W1l   = (const float*)d_in[2];
  const float* W1r   = (const float*)d_in[3];
  const float* att1  = (const float*)d_in[4];
  const float* b1    = (const float*)d_in[5];
  const float* gamma = (const float*)d_in[6];
  const float* beta  = (const float*)d_in[7];
  const float* W2l   = (const float*)d_in[8];
  const float* W2r   = (const float*)d_in[9];
  const float* att2  = (const float*)d_in[10];
  const float* b2    = (const float*)d_in[11];
  float* out = (float*)d_out;

  // Workspace layout (~184 MB)
  float* xl1  = (float*)d_ws;                        // N*128
  float* xr1  = xl1 + (size_t)NNODES * HC1;          // N*128
  float* hbuf = xr1 + (size_t)NNODES * HC1;          // N*128
  float* elog = hbuf + (size_t)NNODES * HC1;         // ETOT*4
  unsigned int* mkey = (unsigned int*)(elog + (size_t)ETOT * HEADS);  // N*4
  float* denom = (float*)(mkey + (size_t)NNODES * HEADS);             // N*4
  float* bnsum = denom + (size_t)NNODES * HEADS;     // 128
  float* bnss  = bnsum + HC1;                        // 128

  dim3 gblk(32, 4);
  const int strips      = NNODES >> 4;               // 6250 16-row strips
  const int stripBlocks = (strips + 3) / 4;
  const int edgeBlocks  = (ETOT + 7) / 8;            // 8 waves/block
  const int ehBlocks    = (ETOT * HEADS + 255) / 256;

  // ---------------- Layer 1 ----------------
  gemm_wmma_f32<F_IN, HC1><<<stripBlocks, gblk, 0, stream>>>(x, W1l, xl1, NNODES);
  gemm_wmma_f32<F_IN, HC1><<<stripBlocks, gblk, 0, stream>>>(x, W1r, xr1, NNODES);

  // zero maxkey + denom (adjacent: 2 * N*4 dwords)
  fill_u32_kernel<<<(2 * NNODES * HEADS + 255) / 256, 256, 0, stream>>>(
      mkey, 0u, 2 * NNODES * HEADS);

  edge_logits_kernel<32><<<edgeBlocks, 256, 0, stream>>>(xl1, xr1, ei, att1, elog, mkey);
  edge_exp_kernel<<<ehBlocks, 256, 0, stream>>>(elog, ei, mkey, denom);

  init_bias_kernel<<<(NNODES * HC1 + 255) / 256, 256, 0, stream>>>(
      hbuf, b1, NNODES * HC1, HC1 - 1);
  edge_agg_kernel<32><<<edgeBlocks, 256, 0, stream>>>(xl1, elog, denom, ei, hbuf);

  // ---------------- BatchNorm + ReLU ----------------
  fill_u32_kernel<<<1, 256, 0, stream>>>((unsigned int*)bnsum, 0u, 2 * HC1);
  bn_reduce_kernel<<<512, 256, 0, stream>>>(hbuf, bnsum, bnss);
  bn_apply_kernel<<<(NNODES * HC1 + 255) / 256, 256, 0, stream>>>(
      hbuf, bnsum, bnss, gamma, beta);

  // ---------------- Layer 2 (reuse xl1/xr1 as hl/hr) ----------------
  float* hl = xl1;
  float* hr = xr1;
  gemm_wmma_f32<HC1, HC2><<<stripBlocks, gblk, 0, stream>>>(hbuf, W2l, hl, NNODES);
  gemm_wmma_f32<HC1, HC2><<<stripBlocks, gblk, 0, stream>>>(hbuf, W2r, hr, NNODES);

  fill_u32_kernel<<<(2 * NNODES * HEADS + 255) / 256, 256, 0, stream>>>(
      mkey, 0u, 2 * NNODES * HEADS);

  edge_logits_kernel<16><<<edgeBlocks, 256, 0, stream>>>(hl, hr, ei, att2, elog, mkey);
  edge_exp_kernel<<<ehBlocks, 256, 0, stream>>>(elog, ei, mkey, denom);

  init_bias_kernel<<<(NNODES * HC2 + 255) / 256, 256, 0, stream>>>(
      out, b2, NNODES * HC2, HC2 - 1);
  edge_agg_kernel<16><<<edgeBlocks, 256, 0, stream>>>(hl, elog, denom, ei, out);
}